// LNN_68126771249709
// MI455X (gfx1250) — compile-verified
//
#include <hip/hip_runtime.h>
#include <math.h>

typedef float v2f __attribute__((ext_vector_type(2)));
typedef float v4f __attribute__((ext_vector_type(4)));
typedef float v8f __attribute__((ext_vector_type(8)));
typedef unsigned int u32x4 __attribute__((ext_vector_type(4)));
typedef int i32x4 __attribute__((ext_vector_type(4)));
typedef int i32x8 __attribute__((ext_vector_type(8)));

#define T_DIM   32768
#define IN_DIM  128
#define R_DIM   1024
#define OUT_DIM 64
#define LEAK    0.5f
#define NBLK    16
#define RPB     (R_DIM / NBLK)   // 64 rows of W per workgroup

// ---------------------------------------------------------------------------
// Kernel 1: U = X @ Win^T   [T x R], fp32 WMMA 16x16x4.
// One wave per 16x16 output tile; K loop = 128/4 = 32 WMMAs.
// ---------------------------------------------------------------------------
__global__ __launch_bounds__(128) void k_in_proj(const float* __restrict__ X,
                                                 const float* __restrict__ Win,
                                                 float* __restrict__ U) {
    const int lane = threadIdx.x & 31;
    const int wave = threadIdx.x >> 5;
    const int t0 = blockIdx.x * 16;
    const int r0 = blockIdx.y * 64 + wave * 16;
    const int mn   = lane & 15;              // M for A, N for B/D
    const int koff = (lane >> 4) << 1;       // lanes 0-15 -> K{0,1}, 16-31 -> K{2,3}
    const float* xrow = X   + (size_t)(t0 + mn) * IN_DIM;
    const float* wrow = Win + (size_t)(r0 + mn) * IN_DIM;   // B[k][n] = Win[r0+n][k]
    v8f acc = {};
    for (int k0 = 0; k0 < IN_DIM; k0 += 4) {
        v2f a = *(const v2f*)(xrow + k0 + koff);
        v2f b = *(const v2f*)(wrow + k0 + koff);
        acc = __builtin_amdgcn_wmma_f32_16x16x4_f32(false, a, false, b,
                                                    (short)0, acc, false, false);
    }
    const int mbase = (lane >> 4) * 8;       // D: VGPR j = row j (lo lanes) / j+8 (hi lanes)
    float* uout = U + (size_t)(t0 + mbase) * R_DIM + r0 + mn;
#pragma unroll
    for (int j = 0; j < 8; ++j)
        uout[(size_t)j * R_DIM] = acc[j];
}

// ---------------------------------------------------------------------------
// Kernel 2: persistent reservoir scan. 16 workgroups; each stages 64 rows of W
// (256 KB, row-major) into LDS via the Tensor Data Mover, then runs 32768
// dependent matvec+tanh steps. Lanes span the K dim (b128 LDS reads,
// conflict-free), per-row sums reduced with a shfl_xor butterfly. State is
// double-buffered in global memory; steps sync on a monotonic atomic barrier.
// ---------------------------------------------------------------------------
__global__ __launch_bounds__(256) void k_reservoir(const float* __restrict__ W,
                                                   float* __restrict__ UST,
                                                   float* __restrict__ S,
                                                   unsigned int* __restrict__ bar) {
    extern __shared__ float lds[];
    float* Wrow = lds;                           // [RPB][R_DIM] row-major slice
    float* sdec = Wrow + (size_t)RPB * R_DIM;    // [R_DIM] decayed state
    float* red  = sdec + R_DIM;                  // [RPB] per-row dot results

    const int tid  = threadIdx.x;
    const int b    = blockIdx.x;
    const int lane = tid & 31;
    const int wave = tid >> 5;

    // --- One-shot TDM stage: W rows [b*RPB, b*RPB+RPB) -> LDS offset 0 ------
    if (wave == 0) {
        unsigned long long ga =
            (unsigned long long)(uintptr_t)(W + (size_t)b * RPB * R_DIM);
        // D# group 0: count=1 | lds_addr=0 | global_addr | type=2 ("image")
        u32x4 g0 = { 1u,
                     0u,
                     (unsigned int)(ga & 0xFFFFFFFFull),
                     (unsigned int)((ga >> 32) & 0x01FFFFFFull) | 0x80000000u };
        // D# group 1: data_size=4B; tensor_dim0=1024; tensor_dim1=1024;
        //             tile_dim0=1024; tile_dim1=64; dim0_stride=1024 elements
        i32x8 g1 = { 0x00020000,   // wg_mask=0, data_size=2 (4B)
                     0x04000000,   // tensor_dim0[15:0]=1024 @bits63:48
                     0x04000000,   // tensor_dim1[15:0]=1024 @bits95:80
                     0x04000000,   // tile_dim0=1024 @bits127:112
                     0x00000040,   // tile_dim1=64 rows; tile_dim2=0
                     0x00000400,   // tensor_dim0_stride=1024 (low 32)
                     0, 0 };
        i32x4 gz4 = { 0, 0, 0, 0 };              // groups 2/3 unused (2-D tensor)
        i32x8 gz8 = { 0, 0, 0, 0, 0, 0, 0, 0 };  // extra descriptor words unused
        __builtin_amdgcn_tensor_load_to_lds(g0, g1, gz4, gz4, gz8, 0);
        __builtin_amdgcn_s_wait_tensorcnt(0);
    }
    __syncthreads();

    const int r0  = wave * 8;                    // this wave's 8 rows
    const float oma = 1.0f - LEAK;

    for (int t = 0; t < T_DIM; ++t) {
        const float* Sin  = S + ((t       & 1) ? R_DIM : 0);
        float*       Sout = S + (((t + 1) & 1) ? R_DIM : 0);

        // s_dec = (1-a) * s_prev (full decayed vector, shared in LDS)
        for (int i = tid; i < R_DIM; i += 256)
            sdec[i] = oma * Sin[i];
        __syncthreads();

        // Per-lane slice of s_dec: 8 x float4, reused across all 8 rows.
        v4f sv[8];
#pragma unroll
        for (int c = 0; c < 8; ++c)
            sv[c] = *(const v4f*)(sdec + c * 128 + lane * 4);

#pragma unroll
        for (int i = 0; i < 8; ++i) {
            const float* wr = Wrow + (size_t)(r0 + i) * R_DIM;
            float acc = 0.f;
#pragma unroll
            for (int c = 0; c < 8; ++c) {
                v4f wv = *(const v4f*)(wr + c * 128 + lane * 4);
                acc = fmaf(wv.x, sv[c].x, acc);
                acc = fmaf(wv.y, sv[c].y, acc);
                acc = fmaf(wv.z, sv[c].z, acc);
                acc = fmaf(wv.w, sv[c].w, acc);
            }
            // wave32 butterfly reduction
            acc += __shfl_xor(acc, 16, 32);
            acc += __shfl_xor(acc,  8, 32);
            acc += __shfl_xor(acc,  4, 32);
            acc += __shfl_xor(acc,  2, 32);
            acc += __shfl_xor(acc,  1, 32);
            if (lane == 0) red[r0 + i] = acc;
        }
        __syncthreads();

        if (tid < RPB) {
            const size_t uoff = (size_t)t * R_DIM + b * RPB + tid;
            float y  = UST[uoff] + red[tid];
            float sd = sdec[b * RPB + tid];
            float sn = fmaf(LEAK, tanhf(y), sd);
            Sout[b * RPB + tid] = sn;     // publish chunk for all blocks
            UST[uoff] = sn;               // record state (reuses u storage)
            __threadfence();              // device-visible before arrive
        }
        __syncthreads();

        if (tid == 0) {                   // monotonic device barrier over NBLK blocks
            __hip_atomic_fetch_add(bar, 1u, __ATOMIC_ACQ_REL, __HIP_MEMORY_SCOPE_AGENT);
            const unsigned int target = (unsigned int)(t + 1) * NBLK;
            while (__hip_atomic_load(bar, __ATOMIC_ACQUIRE, __HIP_MEMORY_SCOPE_AGENT) < target)
                __builtin_amdgcn_s_sleep(2);
        }
        __syncthreads();
    }
}

// ---------------------------------------------------------------------------
// Kernel 3: predictions = states @ Wout  [T x 64], fp32 WMMA 16x16x4.
// ---------------------------------------------------------------------------
__global__ __launch_bounds__(128) void k_out_proj(const float* __restrict__ Sts,
                                                  const float* __restrict__ Wout,
                                                  float* __restrict__ Y) {
    const int lane = threadIdx.x & 31;
    const int wave = threadIdx.x >> 5;
    const int t0 = blockIdx.x * 16;
    const int n0 = wave * 16;
    const int mn   = lane & 15;
    const int koff = (lane >> 4) << 1;
    const float* arow = Sts + (size_t)(t0 + mn) * R_DIM;
    v8f acc = {};
    for (int k0 = 0; k0 < R_DIM; k0 += 4) {
        v2f a = *(const v2f*)(arow + k0 + koff);
        v2f b;
        b.x = Wout[(size_t)(k0 + koff)     * OUT_DIM + n0 + mn];
        b.y = Wout[(size_t)(k0 + koff + 1) * OUT_DIM + n0 + mn];
        acc = __builtin_amdgcn_wmma_f32_16x16x4_f32(false, a, false, b,
                                                    (short)0, acc, false, false);
    }
    const int mbase = (lane >> 4) * 8;
    float* yout = Y + (size_t)(t0 + mbase) * OUT_DIM + n0 + mn;
#pragma unroll
    for (int j = 0; j < 8; ++j)
        yout[(size_t)j * OUT_DIM] = acc[j];
}

// ---------------------------------------------------------------------------
extern "C" void kernel_launch(void* const* d_in, const int* in_sizes, int n_in,
                              void* d_out, int out_size, void* d_ws, size_t ws_size,
                              hipStream_t stream) {
    const float* X    = (const float*)d_in[0];   // [T, IN]
    const float* Win  = (const float*)d_in[1];   // [R, IN]
    const float* W    = (const float*)d_in[2];   // [R, R]
    const float* Wout = (const float*)d_in[3];   // [R, OUT]
    float* Y = (float*)d_out;                    // [T, OUT]

    char* ws = (char*)d_ws;
    unsigned int* bar = (unsigned int*)ws;                         // 256 B slot
    float* S   = (float*)(ws + 256);                               // 2 x R f32
    float* UST = (float*)(ws + 256 + 2 * R_DIM * sizeof(float));   // T x R f32

    // Zero barrier counter + initial state (ws is poisoned, not re-poisoned).
    (void)hipMemsetAsync(ws, 0, 256 + 2 * R_DIM * sizeof(float), stream);

    // 1) input projection (WMMA fp32)
    dim3 g1(T_DIM / 16, R_DIM / 64);
    k_in_proj<<<g1, 128, 0, stream>>>(X, Win, UST);

    // 2) persistent reservoir scan, W slice TDM-staged into LDS (~260 KB/WGP)
    const size_t smem = ((size_t)RPB * R_DIM + R_DIM + RPB) * sizeof(float);
    (void)hipFuncSetAttribute((const void*)k_reservoir,
                              hipFuncAttributeMaxDynamicSharedMemorySize, (int)smem);
    k_reservoir<<<NBLK, 256, smem, stream>>>(W, UST, S, bar);

    // 3) readout projection (WMMA fp32)
    k_out_proj<<<T_DIM / 16, 128, 0, stream>>>(UST, Wout, Y);
}